// GAT_22857815949504
// MI455X (gfx1250) — compile-verified
//
#include <hip/hip_runtime.h>
#include <math.h>

// ---------------- problem constants (match reference) ----------------
#define NNODES  50000
#define NEDGES  1600000
#define NIN     128
#define NHID    64
#define NHEADS  4
#define NOUTC   16
#define NGRAPH  128
#define ETOT    (NEDGES + NNODES)   // edges + self loops
#define NEG_SLOPE 0.2f
#define SM_EPS  1e-16f
#define LN_EPS  1e-5f

typedef float v2f __attribute__((ext_vector_type(2)));
typedef float v8f __attribute__((ext_vector_type(8)));

// ---------------- small utility kernels ----------------
__global__ void fill_f32(float* __restrict__ p, float v, size_t n) {
  size_t i = (size_t)blockIdx.x * blockDim.x + threadIdx.x;
  size_t stride = (size_t)gridDim.x * blockDim.x;
  for (size_t j = i; j < n; j += stride) p[j] = v;
}

__global__ void reduce_sum_f32(const float* __restrict__ x, float* __restrict__ out, size_t n) {
  size_t i = (size_t)blockIdx.x * blockDim.x + threadIdx.x;
  size_t stride = (size_t)gridDim.x * blockDim.x;
  float s = 0.f;
  for (size_t j = i; j < n; j += stride) s += x[j];
  atomicAdd(out, s);
}

// ce1[h] = sum_o We1[h*64+o]*ae1[h*64+o]   (h<4)
// ce2[0] = sum_o We2[o]*ae2[o]
__global__ void compute_ce(const float* __restrict__ We1, const float* __restrict__ ae1,
                           const float* __restrict__ We2, const float* __restrict__ ae2,
                           float* __restrict__ ce1, float* __restrict__ ce2) {
  int h = threadIdx.x;
  if (h < NHEADS) {
    float s = 0.f;
    for (int o = 0; o < NHID; ++o) s += We1[h * NHID + o] * ae1[h * NHID + o];
    ce1[h] = s;
  } else if (h == NHEADS) {
    float s = 0.f;
    for (int o = 0; o < NHID; ++o) s += We2[o] * ae2[o];
    ce2[0] = s;
  }
}

// ---------------- WMMA f32 GEMM: C[M,NC] = A[M,K] @ B[K,NC] ----------------
// wave32; one 16x16 output tile per wave; K stepped by 4 through
// V_WMMA_F32_16X16X4_F32. A-fragment per ISA layout (lanes 0-15: K=0,1 ;
// lanes 16-31: K=2,3). Requires M%16==0, K%4==0, NC%16==0 (true here:
// 50000=3125*16, K in {128,256}, NC in {256,64}).
__global__ void wmma_gemm_f32(const float* __restrict__ A, const float* __restrict__ B,
                              float* __restrict__ C, int K, int NC) {
  int lane = threadIdx.x & 31;
  int wave = threadIdx.x >> 5;                 // 0..7 (blockDim=256)
  int colTile = blockIdx.y * 8 + wave;
  if (colTile * 16 >= NC) return;              // wave-uniform: EXEC stays all-1s
  int rowTile = blockIdx.x;
  int hi = lane >> 4;                          // 0 or 1
  int lo = lane & 15;

  const float* Arow = A + (size_t)(rowTile * 16 + lo) * K;
  const float* Bcol = B + colTile * 16 + lo;

  v8f acc = {};
  for (int kb = 0; kb < K; kb += 4) {
    int ka = kb + 2 * hi;
    v2f a, b;
    a[0] = Arow[ka];
    a[1] = Arow[ka + 1];
    const float* Bp = Bcol + (size_t)ka * NC;
    b[0] = Bp[0];
    b[1] = Bp[NC];
    acc = __builtin_amdgcn_wmma_f32_16x16x4_f32(
        /*neg_a=*/false, a, /*neg_b=*/false, b,
        /*c_mod=*/(short)0, acc, /*reuse_a=*/false, /*reuse_b=*/false);
  }
  // D layout: VGPR j -> row = rowTile*16 + j + 8*hi, col = colTile*16 + lo
  float* Cp = C + (size_t)(rowTile * 16 + 8 * hi) * NC + colTile * 16 + lo;
#pragma unroll
  for (int j = 0; j < 8; ++j) Cp[(size_t)j * NC] = acc[j];
}

// ---------------- attention helpers ----------------
__device__ __forceinline__ void edge_sd(const int* __restrict__ ei, int e, int& s, int& d) {
  if (e < NEDGES) { s = ei[e]; d = ei[NEDGES + e]; }
  else            { s = e - NEDGES; d = s; }            // self loop
}

__device__ __forceinline__ float edge_attr_val(const float* __restrict__ ea,
                                               const float* __restrict__ easum, int e) {
  return (e < NEDGES) ? ea[e] : easum[0] * (1.0f / (float)NEDGES);
}

__device__ __forceinline__ float leaky(float a) { return a > 0.f ? a : NEG_SLOPE * a; }

__device__ __forceinline__ void atomicMaxF(float* addr, float val) {
  unsigned int* ua = (unsigned int*)addr;
  unsigned int old = *ua;
  while (true) {
    float f = __uint_as_float(old);
    if (f >= val) break;
    unsigned int assumed = old;
    old = atomicCAS(ua, assumed, __float_as_uint(val));
    if (old == assumed) break;
  }
}

// per-node attention coefficients: outs[n,h]=h.dot(a_src[h]), outd similarly
__global__ void node_alpha(const float* __restrict__ h, const float* __restrict__ aw_s,
                           const float* __restrict__ aw_d, float* __restrict__ outs,
                           float* __restrict__ outd, int heads) {
  int i = blockIdx.x * blockDim.x + threadIdx.x;
  if (i >= NNODES * heads) return;
  int n = i / heads, hh = i % heads;
  const float* hp = h + (size_t)n * heads * NHID + hh * NHID;
  const float* ws = aw_s + hh * NHID;
  const float* wd = aw_d + hh * NHID;
  float ss = 0.f, sd = 0.f;
  for (int o = 0; o < NHID; ++o) { float v = hp[o]; ss += v * ws[o]; sd += v * wd[o]; }
  outs[i] = ss;
  outd[i] = sd;
}

__global__ void edge_max(const int* __restrict__ ei, const float* __restrict__ ea,
                         const float* __restrict__ easum, const float* __restrict__ ce,
                         const float* __restrict__ asv, const float* __restrict__ adv,
                         float* __restrict__ m, int heads) {
  int i = blockIdx.x * blockDim.x + threadIdx.x;
  if (i >= ETOT * heads) return;
  int e = i / heads, hh = i % heads;
  int s, d; edge_sd(ei, e, s, d);
  float a = leaky(asv[s * heads + hh] + adv[d * heads + hh] + edge_attr_val(ea, easum, e) * ce[hh]);
  atomicMaxF(&m[d * heads + hh], a);
}

__global__ void edge_expsum(const int* __restrict__ ei, const float* __restrict__ ea,
                            const float* __restrict__ easum, const float* __restrict__ ce,
                            const float* __restrict__ asv, const float* __restrict__ adv,
                            const float* __restrict__ m, float* __restrict__ den, int heads) {
  int i = blockIdx.x * blockDim.x + threadIdx.x;
  if (i >= ETOT * heads) return;
  int e = i / heads, hh = i % heads;
  int s, d; edge_sd(ei, e, s, d);
  float a = leaky(asv[s * heads + hh] + adv[d * heads + hh] + edge_attr_val(ea, easum, e) * ce[hh]);
  atomicAdd(&den[d * heads + hh], __expf(a - m[d * heads + hh]));
}

__global__ void edge_message(const int* __restrict__ ei, const float* __restrict__ ea,
                             const float* __restrict__ easum, const float* __restrict__ ce,
                             const float* __restrict__ asv, const float* __restrict__ adv,
                             const float* __restrict__ m, const float* __restrict__ den,
                             const float* __restrict__ h, float* __restrict__ agg, int heads) {
  int i = blockIdx.x * blockDim.x + threadIdx.x;
  if (i >= ETOT * heads) return;
  int e = i / heads, hh = i % heads;
  int s, d; edge_sd(ei, e, s, d);
  float a = leaky(asv[s * heads + hh] + adv[d * heads + hh] + edge_attr_val(ea, easum, e) * ce[hh]);
  float w = __expf(a - m[d * heads + hh]) / (den[d * heads + hh] + SM_EPS);
  const float* hp = h + (size_t)s * heads * NHID + hh * NHID;
  float* ap = agg + (size_t)d * heads * NHID + hh * NHID;
  for (int o = 0; o < NHID; ++o) atomicAdd(&ap[o], hp[o] * w);
}

// buf[n,c] = relu(buf[n,c] + b[c])
__global__ void bias_relu(float* __restrict__ buf, const float* __restrict__ b, int C, size_t total) {
  size_t i = (size_t)blockIdx.x * blockDim.x + threadIdx.x;
  if (i >= total) return;
  int c = (int)(i % C);
  buf[i] = fmaxf(buf[i] + b[c], 0.f);
}

// pooling: pool[g,c] += agg2[n,c]+b2[c]; cnt[g] += 1
__global__ void pool_sum(const float* __restrict__ agg2, const float* __restrict__ b2,
                         const int* __restrict__ batch, float* __restrict__ pool,
                         float* __restrict__ cnt) {
  int i = blockIdx.x * blockDim.x + threadIdx.x;
  if (i >= NNODES * NHID) return;
  int n = i >> 6, c = i & 63;
  int g = batch[n];
  atomicAdd(&pool[g * NHID + c], agg2[i] + b2[c]);
  if (c == 0) atomicAdd(&cnt[g], 1.0f);
}

// per-graph MLP head: Linear(64,64)+ReLU+LayerNorm(64)+Linear(64,16)
__global__ void mlp_head(const float* __restrict__ pool, const float* __restrict__ cnt,
                         const float* __restrict__ Wl1, const float* __restrict__ bl1,
                         const float* __restrict__ lng, const float* __restrict__ lnb,
                         const float* __restrict__ Wl2, const float* __restrict__ bl2,
                         float* __restrict__ out) {
  int g = blockIdx.x;
  int c = threadIdx.x;                      // 64 threads
  __shared__ float gv[NHID], zv[NHID], red[NHID];
  float inv = 1.0f / fmaxf(cnt[g], 1.0f);
  gv[c] = pool[g * NHID + c] * inv;
  __syncthreads();
  float s = bl1[c];
  for (int k = 0; k < NHID; ++k) s += gv[k] * Wl1[k * NHID + c];
  s = fmaxf(s, 0.f);
  red[c] = s;
  __syncthreads();
  for (int off = 32; off > 0; off >>= 1) { if (c < off) red[c] += red[c + off]; __syncthreads(); }
  float mu = red[0] * (1.0f / NHID);
  __syncthreads();
  float dv = (s - mu) * (s - mu);
  red[c] = dv;
  __syncthreads();
  for (int off = 32; off > 0; off >>= 1) { if (c < off) red[c] += red[c + off]; __syncthreads(); }
  float var = red[0] * (1.0f / NHID);
  zv[c] = (s - mu) * rsqrtf(var + LN_EPS) * lng[c] + lnb[c];
  __syncthreads();
  if (c < NOUTC) {
    float o = bl2[c];
    for (int k = 0; k < NHID; ++k) o += zv[k] * Wl2[k * NOUTC + c];
    out[g * NOUTC + c] = o;
  }
}

// ---------------- host-side launcher ----------------
static inline int cdivi(long a, long b) { return (int)((a + b - 1) / b); }

extern "C" void kernel_launch(void* const* d_in, const int* in_sizes, int n_in,
                              void* d_out, int out_size, void* d_ws, size_t ws_size,
                              hipStream_t stream) {
  (void)in_sizes; (void)n_in; (void)out_size; (void)ws_size;

  const float* x    = (const float*)d_in[0];
  const int*   ei   = (const int*)d_in[1];
  const float* ea   = (const float*)d_in[2];
  const int*   bidx = (const int*)d_in[3];
  const float* W1   = (const float*)d_in[4];
  const float* as1  = (const float*)d_in[5];
  const float* ad1  = (const float*)d_in[6];
  const float* We1  = (const float*)d_in[7];
  const float* ae1  = (const float*)d_in[8];
  const float* b1   = (const float*)d_in[9];
  const float* W2   = (const float*)d_in[10];
  const float* as2  = (const float*)d_in[11];
  const float* ad2  = (const float*)d_in[12];
  const float* We2  = (const float*)d_in[13];
  const float* ae2  = (const float*)d_in[14];
  const float* b2   = (const float*)d_in[15];
  const float* Wl1  = (const float*)d_in[16];
  const float* bl1  = (const float*)d_in[17];
  const float* lng  = (const float*)d_in[18];
  const float* lnb  = (const float*)d_in[19];
  const float* Wl2  = (const float*)d_in[20];
  const float* bl2  = (const float*)d_in[21];
  float* out = (float*)d_out;

  // ---- workspace layout (floats), ~132 MB ----
  float* ws = (float*)d_ws;
  float* h1    = ws; ws += (size_t)NNODES * NHEADS * NHID;   // [N,256]
  float* hcat  = ws; ws += (size_t)NNODES * NHEADS * NHID;   // agg1 -> relu(.+b1)
  float* h2p   = ws; ws += (size_t)NNODES * NHID;            // [N,64]
  float* agg2  = ws; ws += (size_t)NNODES * NHID;
  float* as1v  = ws; ws += (size_t)NNODES * NHEADS;
  float* ad1v  = ws; ws += (size_t)NNODES * NHEADS;
  float* m1    = ws; ws += (size_t)NNODES * NHEADS;
  float* d1    = ws; ws += (size_t)NNODES * NHEADS;
  float* as2v  = ws; ws += (size_t)NNODES;
  float* ad2v  = ws; ws += (size_t)NNODES;
  float* m2    = ws; ws += (size_t)NNODES;
  float* d2    = ws; ws += (size_t)NNODES;
  float* pool  = ws; ws += (size_t)NGRAPH * NHID;
  float* cnt   = ws; ws += (size_t)NGRAPH;
  float* easum = ws; ws += 1;
  float* ce1   = ws; ws += NHEADS;
  float* ce2   = ws; ws += 1;

  const int TB = 256;
  const int FG = 2048;  // grid-stride fill blocks

  // ---- init ----
  fill_f32<<<FG, TB, 0, stream>>>(hcat, 0.f, (size_t)NNODES * NHEADS * NHID);
  fill_f32<<<FG, TB, 0, stream>>>(agg2, 0.f, (size_t)NNODES * NHID);
  fill_f32<<<FG, TB, 0, stream>>>(m1, -INFINITY, (size_t)NNODES * NHEADS);
  fill_f32<<<FG, TB, 0, stream>>>(d1, 0.f, (size_t)NNODES * NHEADS);
  fill_f32<<<FG, TB, 0, stream>>>(m2, -INFINITY, (size_t)NNODES);
  fill_f32<<<FG, TB, 0, stream>>>(d2, 0.f, (size_t)NNODES);
  fill_f32<<<2, TB, 0, stream>>>(pool, 0.f, (size_t)NGRAPH * NHID);
  fill_f32<<<1, TB, 0, stream>>>(cnt, 0.f, (size_t)NGRAPH);
  fill_f32<<<1, 32, 0, stream>>>(easum, 0.f, (size_t)1);

  // ---- scalar precompute ----
  reduce_sum_f32<<<1024, TB, 0, stream>>>(ea, easum, (size_t)NEDGES);
  compute_ce<<<1, 8, 0, stream>>>(We1, ae1, We2, ae2, ce1, ce2);

  // ---- layer 1 ----
  wmma_gemm_f32<<<dim3(NNODES / 16, 2), TB, 0, stream>>>(x, W1, h1, NIN, NHEADS * NHID);
  node_alpha<<<cdivi((long)NNODES * NHEADS, TB), TB, 0, stream>>>(h1, as1, ad1, as1v, ad1v, NHEADS);
  {
    int g = cdivi((long)ETOT * NHEADS, TB);
    edge_max    <<<g, TB, 0, stream>>>(ei, ea, easum, ce1, as1v, ad1v, m1, NHEADS);
    edge_expsum <<<g, TB, 0, stream>>>(ei, ea, easum, ce1, as1v, ad1v, m1, d1, NHEADS);
    edge_message<<<g, TB, 0, stream>>>(ei, ea, easum, ce1, as1v, ad1v, m1, d1, h1, hcat, NHEADS);
  }
  bias_relu<<<cdivi((long)NNODES * NHEADS * NHID, TB), TB, 0, stream>>>(
      hcat, b1, NHEADS * NHID, (size_t)NNODES * NHEADS * NHID);

  // ---- layer 2 ----
  wmma_gemm_f32<<<dim3(NNODES / 16, 1), TB, 0, stream>>>(hcat, W2, h2p, NHEADS * NHID, NHID);
  node_alpha<<<cdivi((long)NNODES, TB), TB, 0, stream>>>(h2p, as2, ad2, as2v, ad2v, 1);
  {
    int g = cdivi((long)ETOT, TB);
    edge_max    <<<g, TB, 0, stream>>>(ei, ea, easum, ce2, as2v, ad2v, m2, 1);
    edge_expsum <<<g, TB, 0, stream>>>(ei, ea, easum, ce2, as2v, ad2v, m2, d2, 1);
    edge_message<<<g, TB, 0, stream>>>(ei, ea, easum, ce2, as2v, ad2v, m2, d2, h2p, agg2, 1);
  }

  // ---- pooling + MLP head ----
  pool_sum<<<cdivi((long)NNODES * NHID, TB), TB, 0, stream>>>(agg2, b2, bidx, pool, cnt);
  mlp_head<<<NGRAPH, NHID, 0, stream>>>(pool, cnt, Wl1, bl1, lng, lnb, Wl2, bl2, out);

  // ---- tuple output: second element is edge_attr passthrough ----
  hipMemcpyAsync(out + NGRAPH * NOUTC, ea, (size_t)NEDGES * sizeof(float),
                 hipMemcpyDeviceToDevice, stream);
}